// ResFCN_19533511262823
// MI455X (gfx1250) — compile-verified
//
#include <hip/hip_runtime.h>
#include <hip/hip_bf16.h>

// Problem constants (from reference): B=16, N=32, H=W=480
#define B_    16
#define N_    32
#define H_    480
#define W_    480
#define HW_   (H_ * W_)          // 230400 floats per (b) plane
#define SLABS 90                 // spatial slabs per batch
#define SLABF (HW_ / SLABS)      // 2560 floats per slab
#define STEPS (SLABF / 32)       // 80 steps: 8 waves x 4 floats per step

typedef float v2f __attribute__((ext_vector_type(2)));
typedef float v8f __attribute__((ext_vector_type(8)));

// ---------------------------------------------------------------------------
// Kernel 1: per (b, slab) compute partial overlap[n] (n=0..31) and partial
// target-area using V_WMMA_F32_16X16X4_F32.
//   A (16x4, MxK): M = object, K = 4 consecutive spatial elements
//   B (4x16, KxN): target value broadcast to all 16 columns
//   C (16x16 f32): every column holds the per-object partial dot product
// ISA VGPR layout (CDNA5 ISA):
//   A: lanes 0-15 hold M=lane, VGPR0=K0,VGPR1=K1; lanes 16-31 hold K2,K3
//   B: lanes 0-15 hold N=lane, VGPR0=K0,VGPR1=K1; lanes 16-31 hold K2,K3
//   C: lane(0-15) VGPRr = C[M=r][N=lane]; lane(16-31) VGPRr = C[M=8+r][...]
// ---------------------------------------------------------------------------
__global__ __launch_bounds__(256) void edge_overlap_wmma(
    const float* __restrict__ tgt,   // (B, H, W)
    const float* __restrict__ obj,   // (B, N, H, W)
    float* __restrict__ part)        // (B, SLABS, 33): [0..31]=overlap, [32]=tgt area
{
    const int blk  = blockIdx.x;
    const int b    = blk / SLABS;
    const int slab = blk % SLABS;
    const int tid  = threadIdx.x;
    const int wave = tid >> 5;       // 0..7
    const int lane = tid & 31;
    const int half = lane >> 4;      // 0: K=0,1   1: K=2,3
    const int m    = lane & 15;      // object row within 16-group

    const float* __restrict__ tb = tgt + (size_t)b * HW_;
    const float* __restrict__ ob = obj + (size_t)b * N_ * HW_;
    const float* __restrict__ a0row = ob + (size_t)m * HW_;         // objects 0..15
    const float* __restrict__ a1row = ob + (size_t)(m + 16) * HW_;  // objects 16..31

    v8f acc0 = {};   // objects 0..15
    v8f acc1 = {};   // objects 16..31
    float tsum = 0.0f;

    // this lane's spatial base: slab start + wave chunk + K-half offset
    const int pbase = slab * SLABF + wave * 4 + half * 2;

#pragma unroll 4
    for (int s = 0; s < STEPS; ++s) {
        const int p = pbase + s * 32;                 // 8 waves x 4 floats / step
        const v2f av0 = *(const v2f*)(a0row + p);     // global_load_b64
        const v2f av1 = *(const v2f*)(a1row + p);
        const v2f tv  = *(const v2f*)(tb + p);        // shared by both WMMAs

        // D = A x B + C   (v_wmma_f32_16x16x4_f32)
        acc0 = __builtin_amdgcn_wmma_f32_16x16x4_f32(
            false, av0, false, tv, (short)0, acc0, false, false);
        acc1 = __builtin_amdgcn_wmma_f32_16x16x4_f32(
            false, av1, false, tv, (short)0, acc1, false, false);

        // lanes 0 and 16 together cover K=0..3 exactly once
        tsum += tv.x + tv.y;
    }

    // Block reduction: each wave contributes 32 object partials + 2 tgt halves.
    __shared__ float red[8 * 34];
    if ((lane & 15) == 0) {          // lanes 0 and 16 hold column N=0 / N=16? -> N=lane-16*half, any column is valid (broadcast B)
#pragma unroll
        for (int r = 0; r < 8; ++r) {
            red[wave * 34 + half * 8 + r]      = acc0[r];   // objects half*8 + r
            red[wave * 34 + 16 + half * 8 + r] = acc1[r];   // objects 16 + half*8 + r
        }
        red[wave * 34 + 32 + half] = tsum;    // tgt partial (K halves)
    }
    __syncthreads();

    if (tid < 33) {
        float s = 0.0f;
        if (tid < 32) {
#pragma unroll
            for (int w = 0; w < 8; ++w) s += red[w * 34 + tid];
        } else {
#pragma unroll
            for (int w = 0; w < 8; ++w) s += red[w * 34 + 32] + red[w * 34 + 33];
        }
        part[(size_t)(b * SLABS + slab) * 33 + tid] = s;
    }
}

// ---------------------------------------------------------------------------
// Kernel 2: one block per (b,n). Sums target over the bbox (intersection),
// reduces slab partials (overlap, tgt_area), computes IoU, writes (B,N,2).
// ---------------------------------------------------------------------------
__global__ __launch_bounds__(256) void edge_iou(
    const float* __restrict__ tgt,     // (B, H, W)
    const int*  __restrict__ bboxes,   // (B, N, 4) x1,y1,x2,y2
    const float* __restrict__ part,    // (B, SLABS, 33)
    float* __restrict__ out)           // (B, N, 2)
{
    const int b   = blockIdx.x >> 5;
    const int n   = blockIdx.x & 31;
    const int tid = threadIdx.x;

    const int* bb = bboxes + (size_t)(b * N_ + n) * 4;
    const int x1 = bb[0], y1 = bb[1], x2 = bb[2], y2 = bb[3];

    const float* __restrict__ tb = tgt + (size_t)b * HW_;

    // intersection: waves stride rows, lanes stride columns (coalesced)
    float inter = 0.0f;
    for (int y = y1 + (tid >> 5); y < y2; y += 8) {
        const float* row = tb + y * W_;
        for (int x = x1 + (tid & 31); x < x2; x += 32) inter += row[x];
    }

    // reduce slab partials for overlap[b,n] and tgt_area[b]
    float ov = 0.0f, ta = 0.0f;
    for (int s = tid; s < SLABS; s += 256) {
        const float* ps = part + (size_t)(b * SLABS + s) * 33;
        ov += ps[n];
        ta += ps[32];
    }

    __shared__ float r0[256], r1[256], r2[256];
    r0[tid] = inter; r1[tid] = ov; r2[tid] = ta;
    __syncthreads();
#pragma unroll
    for (int off = 128; off > 0; off >>= 1) {
        if (tid < off) {
            r0[tid] += r0[tid + off];
            r1[tid] += r1[tid + off];
            r2[tid] += r2[tid + off];
        }
        __syncthreads();
    }

    if (tid == 0) {
        const float bw = (x2 > x1) ? (float)(x2 - x1) : 0.0f;
        const float bh = (y2 > y1) ? (float)(y2 - y1) : 0.0f;
        const float box_area = bw * bh;
        const float inter_s  = r0[0];
        const float uni      = box_area + r2[0] - inter_s;
        const float iou      = inter_s / (uni + 1e-8f);
        out[(size_t)(b * N_ + n) * 2 + 0] = r1[0];   // overlap
        out[(size_t)(b * N_ + n) * 2 + 1] = iou;     // iou
    }
}

extern "C" void kernel_launch(void* const* d_in, const int* in_sizes, int n_in,
                              void* d_out, int out_size, void* d_ws, size_t ws_size,
                              hipStream_t stream) {
    const float* tgt    = (const float*)d_in[0];   // (B,H,W) f32
    const float* obj    = (const float*)d_in[1];   // (B,N,H,W) f32
    const int*   bboxes = (const int*)d_in[2];     // (B,N,4) i32
    float*       out    = (float*)d_out;           // (B,N,2) f32
    float*       part   = (float*)d_ws;            // (B,SLABS,33) f32 partials (~190 KB)

    edge_overlap_wmma<<<B_ * SLABS, 256, 0, stream>>>(tgt, obj, part);
    edge_iou<<<B_ * N_, 256, 0, stream>>>(tgt, bboxes, part, out);
}